// DendriticLayer_50878182588987
// MI455X (gfx1250) — compile-verified
//
#include <hip/hip_runtime.h>

typedef __attribute__((ext_vector_type(2))) float v2f;
typedef __attribute__((ext_vector_type(8))) float v8f;
typedef __attribute__((ext_vector_type(4))) unsigned int u32x4;
typedef __attribute__((ext_vector_type(8))) int i32x8;
typedef __attribute__((ext_vector_type(4))) int i32x4;

namespace {
constexpr int kInDim   = 1024;
constexpr int kNeurons = 2048;
constexpr int kDend    = 16;
constexpr int kNSC     = kNeurons * kDend;    // 32768
constexpr float kSlope = 0.1f;

constexpr int kKC    = 256;                   // K-chunk (dwords per Wd row chunk)
constexpr int kNC    = kInDim / kKC;          // 4 chunks
constexpr int kPitch = kKC + 4;               // LDS row pitch (words): TDM pad +4 dwords
}                                             // -> bank = (4*col + k) % 64, conflict-free

__device__ __forceinline__ float leaky(float v) { return v >= 0.0f ? v : kSlope * v; }

__device__ __forceinline__ v8f wmma_f32(v2f a, v2f b, v8f c) {
    return __builtin_amdgcn_wmma_f32_16x16x4_f32(false, a, false, b, (short)0, c,
                                                 false, false);
}

// XOR-butterfly sum over 16 lanes (masks never touch bit4, halves stay separate).
__device__ __forceinline__ float xor_add16(float p) {
    p += __int_as_float(__builtin_amdgcn_ds_swizzle(__float_as_int(p), 0x041F)); // xor 1
    p += __int_as_float(__builtin_amdgcn_ds_swizzle(__float_as_int(p), 0x081F)); // xor 2
    p += __int_as_float(__builtin_amdgcn_ds_swizzle(__float_as_int(p), 0x101F)); // xor 4
    p += __int_as_float(__builtin_amdgcn_ds_swizzle(__float_as_int(p), 0x201F)); // xor 8
    return p;
}

// TDM: DMA a 16-row x 256-dword tile of Wd (row pitch 1024 floats in memory)
// into LDS at lds_off, inserting 16B padding after every 256 dwords so the
// LDS pitch is kPitch words. All descriptor fields are workgroup-uniform.
// D# group0/group1 per CDNA5 ISA; groups 2/3 are
// only consumed for >2D tensors (tile_dim2 == 0 here), so zeros are correct.
__device__ __forceinline__ void tdm_load_wd_chunk(const float* gtile, unsigned lds_off) {
    unsigned long long ga = (unsigned long long)(uintptr_t)gtile;
    u32x4 g0;
    g0[0] = 1u;                                            // count=1, no gather
    g0[1] = lds_off;                                       // lds_addr (bytes)
    g0[2] = (unsigned)(ga & 0xffffffffu);                  // global_addr[31:0]
    g0[3] = (unsigned)((ga >> 32) & 0x01ffffffu)           // global_addr[56:32]
          | 0x80000000u;                                   // type=2 ("image")
    i32x8 g1;
    g1[0] = (2 << 16)      // data_size = 4 bytes
          | (1 << 20)      // pad_enable
          | (7 << 22)      // pad_interval: 256 dwords
          | (3 << 25);     // pad_amount:   4 dwords (16B)
    g1[1] = (int)((unsigned)kKC << 16);                    // tensor_dim0[15:0] = 256
    g1[2] = (int)(16u << 16);                              // tensor_dim1[15:0] = 16
    g1[3] = (int)((unsigned)kKC << 16);                    // tile_dim0 = 256
    g1[4] = 16;                                            // tile_dim1 = 16, tile_dim2 = 0
    g1[5] = kInDim;                                        // tensor_dim0_stride = 1024
    g1[6] = 0;
    g1[7] = 0;
    i32x4 z4 = {0, 0, 0, 0};
    i32x8 z8 = {0, 0, 0, 0, 0, 0, 0, 0};
    __builtin_amdgcn_tensor_load_to_lds(g0, g1, z4, z4, z8, 0);
}

// One workgroup per neuron (2048 blocks x 256 threads = 8 waves).
// TDM double-buffers 16x256 Wd chunks into LDS; each wave runs two 16x16
// batch tiles with v_wmma_f32_16x16x4_f32 (B read from LDS, shared by both
// tiles), software-pipelined one K-step ahead, then fuses the block-diagonal
// soma reduction (bd folded into acc init, ds_swizzle butterfly, bs+leaky).
__global__ __launch_bounds__(256)
void dendritic_fused_wmma(const float* __restrict__ x,
                          const float* __restrict__ Wd,
                          const float* __restrict__ bd,
                          const float* __restrict__ Ws,
                          const float* __restrict__ bs,
                          float* __restrict__ y)
{
    __shared__ float bwd[2][16 * kPitch];                  // 2 x 16.25 KB

    const int n     = blockIdx.x;
    const int j0    = n * kDend;
    const int tid   = threadIdx.x;
    const int lane  = tid & 31;
    const int wave  = tid >> 5;          // 0..7
    const int col   = lane & 15;         // N (dendrite) in C-tile; M row for A loads
    const int khalf = lane >> 4;         // 0/1 -> K offset 2*khalf (f32 WMMA layout)

    // Fused second-stage per-lane constants (col == dendrite index d).
    const float bdv = bd[j0 + col];
    const float wsv = Ws[(size_t)n * kNSC + j0 + col];     // block-diagonal element
    const float bsv = bs[n];

    const float* a0row = x + (size_t)(wave * 16 + col) * kInDim + 2 * khalf;
    const float* a1row = x + (size_t)((wave + 8) * 16 + col) * kInDim + 2 * khalf;
    const float* wtile = Wd + (size_t)j0 * kInDim;         // Wd rows of this neuron
    const unsigned lds_base = (unsigned)(uintptr_t)&bwd[0][0];
    constexpr unsigned kBufBytes = 16u * kPitch * 4u;

    if (wave == 0) {                                       // single issuing wave
        tdm_load_wd_chunk(wtile + 0 * kKC, lds_base + 0 * kBufBytes);
        tdm_load_wd_chunk(wtile + 1 * kKC, lds_base + 1 * kBufBytes);
    }

    // bd is per-column and constant over rows: fold it into one split-K acc.
    v8f acc00 = {bdv, bdv, bdv, bdv, bdv, bdv, bdv, bdv};
    v8f acc10 = acc00;
    v8f acc01 = {0.f, 0.f, 0.f, 0.f, 0.f, 0.f, 0.f, 0.f};
    v8f acc11 = acc01;

    #pragma unroll
    for (int c = 0; c < kNC; ++c) {
        if (wave == 0) {
            // TDM completes in-order: <=1 outstanding => chunk c has landed.
            if (c == kNC - 1) __builtin_amdgcn_s_wait_tensorcnt(0);
            else              __builtin_amdgcn_s_wait_tensorcnt(1);
        }
        __syncthreads();                                   // buf[c&1] visible to all

        const float* aw0 = a0row + c * kKC;
        const float* aw1 = a1row + c * kKC;
        const float* bl  = &bwd[c & 1][col * kPitch + 2 * khalf];

        // Software pipeline: fetch K-step k+8 before consuming K-step k, so
        // >=4 WMMAs separate every load from its use (no loadcnt-0 drains).
        v2f b0  = *(const v2f*)(bl);
        v2f b1  = *(const v2f*)(bl + 4);
        v2f a00 = *(const v2f*)(aw0);
        v2f a01 = *(const v2f*)(aw0 + 4);
        v2f a10 = *(const v2f*)(aw1);
        v2f a11 = *(const v2f*)(aw1 + 4);

        #pragma unroll 4
        for (int k = 8; k < kKC; k += 8) {
            v2f nb0  = *(const v2f*)(bl + k);              // ds_load_2addr_b64 pair
            v2f nb1  = *(const v2f*)(bl + k + 4);
            v2f na00 = *(const v2f*)(aw0 + k);
            v2f na01 = *(const v2f*)(aw0 + k + 4);
            v2f na10 = *(const v2f*)(aw1 + k);
            v2f na11 = *(const v2f*)(aw1 + k + 4);
            acc00 = wmma_f32(a00, b0, acc00);
            acc01 = wmma_f32(a01, b1, acc01);
            acc10 = wmma_f32(a10, b0, acc10);
            acc11 = wmma_f32(a11, b1, acc11);
            b0 = nb0; b1 = nb1;
            a00 = na00; a01 = na01; a10 = na10; a11 = na11;
        }
        acc00 = wmma_f32(a00, b0, acc00);                  // drain last K-step
        acc01 = wmma_f32(a01, b1, acc01);
        acc10 = wmma_f32(a10, b0, acc10);
        acc11 = wmma_f32(a11, b1, acc11);

        __syncthreads();                                   // all waves done with buf[c&1]
        if (wave == 0 && c + 2 < kNC) {
            tdm_load_wd_chunk(wtile + (c + 2) * kKC, lds_base + (c & 1) * kBufBytes);
        }
    }

    // Epilogue: h = leaky(acc) (bd already folded in); y = leaky(sum_d h*ws + bs).
    // C layout: VGPR v, lanes 0-15 -> row m0+v, lanes 16-31 -> row m0+v+8,
    // column (dendrite) = lane&15. Reduce across the 16 columns via ds_swizzle.
    #pragma unroll
    for (int v = 0; v < 8; ++v) {
        float p0 = xor_add16(leaky(acc00[v] + acc01[v]) * wsv);
        float p1 = xor_add16(leaky(acc10[v] + acc11[v]) * wsv);
        if (col == 0) {                                    // lanes 0 and 16
            const int r0 = wave * 16 + v + 8 * khalf;
            const int r1 = (wave + 8) * 16 + v + 8 * khalf;
            y[(size_t)r0 * kNeurons + n] = leaky(p0 + bsv);
            y[(size_t)r1 * kNeurons + n] = leaky(p1 + bsv);
        }
    }
}

extern "C" void kernel_launch(void* const* d_in, const int* in_sizes, int n_in,
                              void* d_out, int out_size, void* d_ws, size_t ws_size,
                              hipStream_t stream) {
    const float* x  = (const float*)d_in[0];
    const float* Wd = (const float*)d_in[1];
    const float* bd = (const float*)d_in[2];
    const float* Ws = (const float*)d_in[3];
    const float* bs = (const float*)d_in[4];
    float* y = (float*)d_out;
    (void)in_sizes; (void)n_in; (void)out_size; (void)d_ws; (void)ws_size;

    dendritic_fused_wmma<<<dim3(kNeurons), dim3(256), 0, stream>>>(x, Wd, bd, Ws, bs, y);
}